// GraphUpSamplingLayer_76716705841225
// MI455X (gfx1250) — compile-verified
//
#include <hip/hip_runtime.h>
#include <hip/hip_bf16.h>

typedef __attribute__((ext_vector_type(2))) float v2f;
typedef __attribute__((ext_vector_type(8))) float v8f;

constexpr int B = 4;
constexpr int C = 512;     // 2^9
constexpr int M = 2048;
constexpr int N = 8192;    // 2^13

// ---------------------------------------------------------------------------
// Kernel 1: nearest coarse point per dense point.
// score[n,m] = -2 p_n . s_m + ||s_m||^2   (same argmin as full sq-distance)
// Computed as a 16x16x4 fp32 WMMA:  A row = [px,py,pz,1], B col = [-2sx,-2sy,-2sz,|s|^2]
// Block: 256 threads (8 waves). blockIdx.y = batch, blockIdx.x = 128-row chunk.
// Each wave owns one 16-row tile and sweeps all M in 16-col WMMA tiles.
// ---------------------------------------------------------------------------
__global__ __launch_bounds__(256) void nn_argmin_kernel(
    const float* __restrict__ sub_pos,   // [B, M, 3]
    const float* __restrict__ pos,       // [B, N, 3]
    int* __restrict__ idx_out)           // [B, N]
{
    // Per-m score vector {-2sx,-2sy,-2sz,|s|^2}, 16B stride -> 32KB LDS
    __shared__ float sB[M * 4];

    const int b = blockIdx.y;

    for (int m = threadIdx.x; m < M; m += blockDim.x) {
        const float* sp = sub_pos + ((size_t)b * M + m) * 3;
        const float sx = sp[0], sy = sp[1], sz = sp[2];
        sB[m * 4 + 0] = -2.0f * sx;
        sB[m * 4 + 1] = -2.0f * sy;
        sB[m * 4 + 2] = -2.0f * sz;
        sB[m * 4 + 3] = sx * sx + sy * sy + sz * sz;
    }
    __syncthreads();

    const int lane = threadIdx.x & 31;
    const int wave = threadIdx.x >> 5;
    const int col  = lane & 15;   // column (m within tile) / row-of-A (n within tile)
    const int hi16 = lane >> 4;   // which 16-lane half
    const int rowBase = blockIdx.x * 128 + wave * 16;

    // A matrix (16x4 f32): lanes 0-15 hold row=col with K={0,1}; lanes 16-31 K={2,3}.
    // K layout per row: [px, py, pz, 1.0]
    const float* pp = pos + ((size_t)b * N + (size_t)(rowBase + col)) * 3;
    v2f a;
    a.x = pp[2 * hi16];            // K0 (px) or K2 (pz)
    a.y = hi16 ? 1.0f : pp[1];     // K3 (1)  or K1 (py)

    float bestV[8];
    int   bestI[8];
#pragma unroll
    for (int r = 0; r < 8; ++r) { bestV[r] = __builtin_inff(); bestI[r] = 0; }

#pragma unroll 4
    for (int t = 0; t < M / 16; ++t) {
        const int mbase = t * 16;
        const int mcol  = mbase + col;

        // B matrix (4x16 f32): lanes 0-15: K={0,1}; lanes 16-31: K={2,3}; N=col.
        // Adjacent in LDS -> single 8-byte ds load.
        const float2 bb = *(const float2*)&sB[(size_t)mcol * 4 + hi16 * 2];
        v2f bv; bv.x = bb.x; bv.y = bb.y;

        v8f cc = {};
        cc = __builtin_amdgcn_wmma_f32_16x16x4_f32(
            /*neg_a=*/false, a, /*neg_b=*/false, bv,
            /*c_mod=*/(short)0, cc, /*reuse_a=*/false, /*reuse_b=*/false);

        // C/D layout: lanes 0-15 -> row r, lanes 16-31 -> row r+8, column = col.
#pragma unroll
        for (int r = 0; r < 8; ++r) {
            const float v = cc[r];
            const bool better = v < bestV[r];
            bestV[r] = better ? v : bestV[r];
            bestI[r] = better ? mcol : bestI[r];
        }
    }

    // Reduce across the 16 columns. XOR masks 1/2/4/8 never cross the 16-lane
    // halves, so rows 0-7 (lanes 0-15) and rows 8-15 (lanes 16-31) reduce
    // independently. Tie-break on lower m to match jnp.argmin.
#pragma unroll
    for (int r = 0; r < 8; ++r) {
        float v = bestV[r];
        int   i = bestI[r];
#pragma unroll
        for (int off = 8; off >= 1; off >>= 1) {
            const float ov = __shfl_xor(v, off, 32);
            const int   oi = __shfl_xor(i, off, 32);
            if (ov < v || (ov == v && oi < i)) { v = ov; i = oi; }
        }
        bestV[r] = v;
        bestI[r] = i;
    }

    if (col == 0) {
#pragma unroll
        for (int r = 0; r < 8; ++r) {
            idx_out[(size_t)b * N + rowBase + hi16 * 8 + r] = bestI[r];
        }
    }
}

// ---------------------------------------------------------------------------
// Kernel 2: feature gather. out[b,c,n] = sub_x[b,c,idx[b,n]].
// Write-stream bound (64 MiB); sub_x (16 MiB) is L2-resident. NT store output.
// ---------------------------------------------------------------------------
__global__ __launch_bounds__(256) void gather_kernel(
    const float* __restrict__ sub_x,   // [B, C, M]
    const int*  __restrict__ idx,      // [B, N]
    float* __restrict__ out)           // [B, C, N]
{
    const size_t tid = (size_t)blockIdx.x * blockDim.x + threadIdx.x;
    const int    n   = (int)(tid & (size_t)(N - 1));
    const size_t bc  = tid >> 13;             // / N
    const int    c   = (int)(bc & (C - 1));
    const int    b   = (int)(bc >> 9);        // / C

    const int   m = idx[(size_t)b * N + n];
    const float v = sub_x[((size_t)b * C + c) * (size_t)M + (size_t)m];
    __builtin_nontemporal_store(v, &out[tid]);
}

// ---------------------------------------------------------------------------
extern "C" void kernel_launch(void* const* d_in, const int* in_sizes, int n_in,
                              void* d_out, int out_size, void* d_ws, size_t ws_size,
                              hipStream_t stream) {
    const float* sub_x   = (const float*)d_in[0];  // [B, C, M]
    const float* sub_pos = (const float*)d_in[1];  // [B, M, 3]
    const float* pos     = (const float*)d_in[2];  // [B, N, 3]
    float* out = (float*)d_out;                    // [B, C, N]
    int*   idx = (int*)d_ws;                       // [B, N] scratch (128 KB)

    dim3 grid1(N / 128, B);
    nn_argmin_kernel<<<grid1, 256, 0, stream>>>(sub_pos, pos, idx);

    const size_t total = (size_t)B * C * N;        // 16,777,216
    gather_kernel<<<(unsigned)((total + 255) / 256), 256, 0, stream>>>(sub_x, idx, out);
}